// FFN_40166534152786
// MI455X (gfx1250) — compile-verified
//
#include <hip/hip_runtime.h>
#include <hip/hip_bf16.h>

// ---------------------------------------------------------------------------
// MoE FFN (DeepSeek-style) for gfx1250 (MI455X), fp32, V_WMMA_F32_16X16X4_F32.
//   B*T = 1024 tokens, D = 1024, E = 64 experts, F = 704, top-6, shared Fs=1408
// ---------------------------------------------------------------------------

#define D_DIM   1024
#define F_DIM   704
#define FS_DIM  1408
#define E_NUM   64
#define N_TOK   1024
#define TOPK    6

// LDS row strides padded so stride % 64 == 4  -> column reads hit 16 distinct banks
#define XS_STRIDE   (D_DIM + 4)    // 1028 (4112 B, 16B-divisible -> b128 OK)
#define HS_STRIDE   (F_DIM + 4)    // 708
#define HSH_STRIDE  (FS_DIM + 4)   // 1412

#if __has_builtin(__builtin_amdgcn_global_load_async_to_lds_b128) && \
    __has_builtin(__builtin_amdgcn_s_wait_asynccnt)
#define USE_ASYNC_LDS 1
#else
#define USE_ASYNC_LDS 0
#endif

typedef float v2f __attribute__((ext_vector_type(2)));
typedef float v8f __attribute__((ext_vector_type(8)));

// D = A(16x4 f32) * B(4x16 f32) + C(16x16 f32)
__device__ __forceinline__ v8f wmma_f32(v2f a, v2f b, v8f c) {
    return __builtin_amdgcn_wmma_f32_16x16x4_f32(
        /*neg_a=*/false, a, /*neg_b=*/false, b,
        /*c_mod=*/(short)0, c, /*reuse_a=*/false, /*reuse_b=*/false);
}

// fast sigmoid: v_exp_f32 + v_add + v_rcp_f32 + v_mul (no IEEE div chain)
__device__ __forceinline__ float sigmoidf_(float v) {
    return __builtin_amdgcn_rcpf(1.0f + __expf(-v));
}

#if USE_ASYNC_LDS
// exact pointee type from the compiler diagnostic:
//   __attribute__((vector_size(4*sizeof(int)))) int  (16-byte int vector)
typedef int avec4i __attribute__((vector_size(4 * sizeof(int))));
typedef __attribute__((address_space(1))) avec4i* gvec_ptr;   // global (AS1)
typedef __attribute__((address_space(3))) avec4i* lvec_ptr;   // LDS    (AS3)

// cache->LDS direct 16B copy, tracked on ASYNCcnt (no VGPR data round-trip)
__device__ __forceinline__ void async_copy16(const float* gsrc, float* ldst) {
    __builtin_amdgcn_global_load_async_to_lds_b128(
        (gvec_ptr)gsrc, (lvec_ptr)ldst, /*offset=*/0, /*cpol=*/0);
}
#endif

// ---------------------------------------------------------------------------
// Kernel 0: zero per-expert counters
// ---------------------------------------------------------------------------
__global__ void moe_init_kernel(int* __restrict__ cnt) {
    int i = threadIdx.x;
    if (i < E_NUM) cnt[i] = 0;
}

// ---------------------------------------------------------------------------
// Kernel 1: gate. One block (64 threads) per token.
//   scores[t,e] = sigmoid(x[t] . w_gate[e]) + gate_b[e]   (written to d_out)
//   serial top-6 + weight normalization -> tk_idx / tk_w
// ---------------------------------------------------------------------------
__global__ __launch_bounds__(64) void moe_gate_kernel(
    const float* __restrict__ x, const float* __restrict__ wg,
    const float* __restrict__ gb,
    float* __restrict__ scores_out,
    int* __restrict__ tk_idx, float* __restrict__ tk_w)
{
    __shared__ float xs[D_DIM];
    __shared__ float sc[E_NUM];
    const int tkn = blockIdx.x;
    const int tid = threadIdx.x;

    const float4* xin = (const float4*)(x + (size_t)tkn * D_DIM);
    for (int i = tid; i < D_DIM / 4; i += 64)
        ((float4*)xs)[i] = xin[i];
    __syncthreads();

    // one expert per thread
    const float* wrow = wg + (size_t)tid * D_DIM;
    float acc = 0.0f;
    for (int k = 0; k < D_DIM; k += 4) {
        float4 w4 = *(const float4*)(wrow + k);
        acc += xs[k + 0] * w4.x + xs[k + 1] * w4.y +
               xs[k + 2] * w4.z + xs[k + 3] * w4.w;
    }
    float s = sigmoidf_(acc) + gb[tid];
    sc[tid] = s;
    scores_out[(size_t)tkn * E_NUM + tid] = s;
    __syncthreads();

    if (tid == 0) {
        float wv[TOPK]; int iv[TOPK]; float wsum = 0.0f;
        for (int kk = 0; kk < TOPK; ++kk) {
            float best = -1e30f; int bi = 0;
            for (int e = 0; e < E_NUM; ++e) {
                float v = sc[e];
                if (v > best) { best = v; bi = e; }
            }
            sc[bi] = -1e30f;            // knock out winner (ties: lowest index)
            wv[kk] = best; iv[kk] = bi; wsum += best;
        }
        float inv = 1.0f / wsum;        // once per token: IEEE div is fine
        for (int kk = 0; kk < TOPK; ++kk) {
            tk_idx[tkn * TOPK + kk] = iv[kk];
            tk_w[tkn * TOPK + kk]   = wv[kk] * inv;
        }
    }
}

// ---------------------------------------------------------------------------
// Kernel 2: scatter tokens into per-expert lists
// ---------------------------------------------------------------------------
__global__ __launch_bounds__(256) void moe_scatter_kernel(
    const int* __restrict__ tk_idx, const float* __restrict__ tk_w,
    int* __restrict__ cnt, int* __restrict__ list, float* __restrict__ lw)
{
    int tkn = blockIdx.x * blockDim.x + threadIdx.x;
    if (tkn >= N_TOK) return;
    for (int kk = 0; kk < TOPK; ++kk) {
        int   e = tk_idx[tkn * TOPK + kk];
        float w = tk_w[tkn * TOPK + kk];
        int pos = atomicAdd(&cnt[e], 1);
        list[e * N_TOK + pos] = tkn;
        lw[e * N_TOK + pos]   = w;
    }
}

// ---------------------------------------------------------------------------
// WMMA fragment helpers (ISA §7.12.2 layouts)
//   A 16x4:  lane m = lane&15, kh = lane>>4 ; a = {X[m][k+2kh], X[m][k+2kh+1]}
//   B 4x16:  lane n = lane&15 ; b = {W[(k+2kh)*ld + nb+n], W[(k+2kh+1)*ld + nb+n]}
//   C/D:     reg r holds (m = r + 8*kh, n = lane&15)
// ---------------------------------------------------------------------------

// ---------------------------------------------------------------------------
// Kernel 3: shared expert SwiGLU — writes out (no prior init needed).
//   One block per 16-token tile. 256 threads = 8 waves.
// ---------------------------------------------------------------------------
__global__ __launch_bounds__(256) void moe_shared_kernel(
    const float* __restrict__ x,
    const float* __restrict__ w1s, const float* __restrict__ w2s,
    const float* __restrict__ w3s, float* __restrict__ out)
{
    extern __shared__ float smem[];
    float* Xs = smem;                       // [16][XS_STRIDE]
    float* Hs = smem + 16 * XS_STRIDE;      // [16][HSH_STRIDE]

    const int t0   = blockIdx.x * 16;
    const int tid  = threadIdx.x;
    const int wave = tid >> 5;
    const int lane = tid & 31;
    const int m    = lane & 15;
    const int kh   = lane >> 4;

    // load 16 token rows into LDS
#if USE_ASYNC_LDS
    for (int i = tid; i < 16 * (D_DIM / 4); i += 256) {
        int row = i / (D_DIM / 4), col4 = i % (D_DIM / 4);
        async_copy16(x + (size_t)(t0 + row) * D_DIM + col4 * 4,
                     Xs + row * XS_STRIDE + col4 * 4);
    }
    __builtin_amdgcn_s_wait_asynccnt(0);
#else
    for (int i = tid; i < 16 * (D_DIM / 4); i += 256) {
        int row = i / (D_DIM / 4), col4 = i % (D_DIM / 4);
        float4 v = ((const float4*)(x + (size_t)(t0 + row) * D_DIM))[col4];
        ((float4*)(Xs + row * XS_STRIDE))[col4] = v;
    }
#endif
    __syncthreads();

    // ---- stage 1: H = silu(X*W1s) * (X*W3s), F-tiles striped across waves ----
    for (int ft = wave; ft < FS_DIM / 16; ft += 8) {
        const int nb = ft * 16;
        v8f c1 = {0,0,0,0,0,0,0,0};
        v8f c3 = {0,0,0,0,0,0,0,0};
#pragma unroll 8
        for (int k = 0; k < D_DIM; k += 4) {
            v2f a;
            a.x = Xs[m * XS_STRIDE + k + 2 * kh];
            a.y = Xs[m * XS_STRIDE + k + 2 * kh + 1];
            size_t r0 = (size_t)(k + 2 * kh) * FS_DIM + nb + m;
            v2f b1; b1.x = w1s[r0]; b1.y = w1s[r0 + FS_DIM];
            v2f b3; b3.x = w3s[r0]; b3.y = w3s[r0 + FS_DIM];
            c1 = wmma_f32(a, b1, c1);
            c3 = wmma_f32(a, b3, c3);
        }
        for (int r = 0; r < 8; ++r) {
            float g  = c1[r];
            float hv = (g * sigmoidf_(g)) * c3[r];
            Hs[(r + 8 * kh) * HSH_STRIDE + nb + m] = hv;
        }
    }
    __syncthreads();

    // ---- stage 2: out = H * W2s (plain store) ----
    for (int nt = wave; nt < D_DIM / 16; nt += 8) {
        const int nb = nt * 16;
        v8f acc = {0,0,0,0,0,0,0,0};
#pragma unroll 8
        for (int k = 0; k < FS_DIM; k += 4) {
            v2f a;
            a.x = Hs[m * HSH_STRIDE + k + 2 * kh];
            a.y = Hs[m * HSH_STRIDE + k + 2 * kh + 1];
            size_t r0 = (size_t)(k + 2 * kh) * D_DIM + nb + m;
            v2f b; b.x = w2s[r0]; b.y = w2s[r0 + D_DIM];
            acc = wmma_f32(a, b, acc);
        }
        for (int r = 0; r < 8; ++r) {
            int row = r + 8 * kh;
            out[(size_t)(t0 + row) * D_DIM + nb + m] = acc[r];
        }
    }
}

// ---------------------------------------------------------------------------
// Kernel 4: routed experts. Grid = (E, N_TOK/16); block handles 16 gathered
// tokens of one expert. Stage 1 -> LDS, stage 2 -> weighted atomic add to out.
// ---------------------------------------------------------------------------
__global__ __launch_bounds__(256) void moe_routed_kernel(
    const float* __restrict__ x,
    const float* __restrict__ w1, const float* __restrict__ w2,
    const float* __restrict__ w3,
    const int* __restrict__ cnt, const int* __restrict__ list,
    const float* __restrict__ lw,
    float* __restrict__ out)
{
    const int e = blockIdx.x;
    const int t = blockIdx.y;
    const int c = cnt[e];
    if (t * 16 >= c) return;                    // uniform: no divergent barrier

    extern __shared__ float smem[];
    float* Xs = smem;                           // [16][XS_STRIDE]
    float* Hs = smem + 16 * XS_STRIDE;          // [16][HS_STRIDE]
    __shared__ int   toks[16];
    __shared__ float wts[16];

    const int tid  = threadIdx.x;
    const int wave = tid >> 5;
    const int lane = tid & 31;
    const int m    = lane & 15;
    const int kh   = lane >> 4;

    if (tid < 16) {
        int row = t * 16 + tid;
        bool v  = row < c;
        toks[tid] = v ? list[e * N_TOK + row] : -1;
        wts[tid]  = v ? lw[e * N_TOK + row] : 0.0f;
    }
    __syncthreads();

    // gather X rows (zeros for padded rows -> zero contribution everywhere)
#if USE_ASYNC_LDS
    for (int i = tid; i < 16 * (D_DIM / 4); i += 256) {
        int row = i / (D_DIM / 4), col4 = i % (D_DIM / 4);
        int tok = toks[row];
        float* ldst = Xs + row * XS_STRIDE + col4 * 4;
        if (tok >= 0) {
            async_copy16(x + (size_t)tok * D_DIM + col4 * 4, ldst);
        } else {
            ((float4*)ldst)[0] = make_float4(0.f, 0.f, 0.f, 0.f);
        }
    }
    __builtin_amdgcn_s_wait_asynccnt(0);
#else
    for (int i = tid; i < 16 * (D_DIM / 4); i += 256) {
        int row = i / (D_DIM / 4), col4 = i % (D_DIM / 4);
        int tok = toks[row];
        float4 v = make_float4(0.f, 0.f, 0.f, 0.f);
        if (tok >= 0) v = ((const float4*)(x + (size_t)tok * D_DIM))[col4];
        ((float4*)(Xs + row * XS_STRIDE))[col4] = v;
    }
#endif
    __syncthreads();

    const float* W1e = w1 + (size_t)e * D_DIM * F_DIM;
    const float* W3e = w3 + (size_t)e * D_DIM * F_DIM;
    const float* W2e = w2 + (size_t)e * F_DIM * D_DIM;

    // ---- stage 1: H = silu(X*W1e) * (X*W3e) ----
    for (int ft = wave; ft < F_DIM / 16; ft += 8) {
        const int nb = ft * 16;
        v8f c1 = {0,0,0,0,0,0,0,0};
        v8f c3 = {0,0,0,0,0,0,0,0};
#pragma unroll 8
        for (int k = 0; k < D_DIM; k += 4) {
            v2f a;
            a.x = Xs[m * XS_STRIDE + k + 2 * kh];
            a.y = Xs[m * XS_STRIDE + k + 2 * kh + 1];
            size_t r0 = (size_t)(k + 2 * kh) * F_DIM + nb + m;
            v2f b1; b1.x = W1e[r0]; b1.y = W1e[r0 + F_DIM];
            v2f b3; b3.x = W3e[r0]; b3.y = W3e[r0 + F_DIM];
            c1 = wmma_f32(a, b1, c1);
            c3 = wmma_f32(a, b3, c3);
        }
        for (int r = 0; r < 8; ++r) {
            float g  = c1[r];
            float hv = (g * sigmoidf_(g)) * c3[r];
            Hs[(r + 8 * kh) * HS_STRIDE + nb + m] = hv;
        }
    }
    __syncthreads();

    // ---- stage 2: out += gate_weight * (H * W2e), atomic combine ----
    for (int nt = wave; nt < D_DIM / 16; nt += 8) {
        const int nb = nt * 16;
        v8f acc = {0,0,0,0,0,0,0,0};
#pragma unroll 8
        for (int k = 0; k < F_DIM; k += 4) {
            v2f a;
            a.x = Hs[m * HS_STRIDE + k + 2 * kh];
            a.y = Hs[m * HS_STRIDE + k + 2 * kh + 1];
            size_t r0 = (size_t)(k + 2 * kh) * D_DIM + nb + m;
            v2f b; b.x = W2e[r0]; b.y = W2e[r0 + D_DIM];
            acc = wmma_f32(a, b, acc);
        }
        for (int r = 0; r < 8; ++r) {
            int row = r + 8 * kh;
            int tok = toks[row];
            if (tok >= 0)
                atomicAdd(&out[(size_t)tok * D_DIM + nb + m], acc[r] * wts[row]);
        }
    }
}

// ---------------------------------------------------------------------------
// Launch
// ---------------------------------------------------------------------------
extern "C" void kernel_launch(void* const* d_in, const int* in_sizes, int n_in,
                              void* d_out, int out_size, void* d_ws, size_t ws_size,
                              hipStream_t stream) {
    const float* x   = (const float*)d_in[0];
    const float* wg  = (const float*)d_in[1];
    const float* gb  = (const float*)d_in[2];
    const float* w1s = (const float*)d_in[3];
    const float* w2s = (const float*)d_in[4];
    const float* w3s = (const float*)d_in[5];
    const float* w1r = (const float*)d_in[6];
    const float* w2r = (const float*)d_in[7];
    const float* w3r = (const float*)d_in[8];

    float* out    = (float*)d_out;                       // [N_TOK, D_DIM]
    float* scores = out + (size_t)N_TOK * D_DIM;         // [N_TOK, E_NUM]

    // workspace layout
    char* ws = (char*)d_ws;
    int*   tk_idx = (int*)ws;                              ws += (size_t)N_TOK * TOPK * sizeof(int);
    float* tk_w   = (float*)ws;                            ws += (size_t)N_TOK * TOPK * sizeof(float);
    int*   cnt    = (int*)ws;                              ws += (size_t)E_NUM * sizeof(int);
    int*   list   = (int*)ws;                              ws += (size_t)E_NUM * N_TOK * sizeof(int);
    float* lw     = (float*)ws;

    moe_init_kernel<<<1, 64, 0, stream>>>(cnt);
    moe_gate_kernel<<<N_TOK, 64, 0, stream>>>(x, wg, gb, scores, tk_idx, tk_w);
    moe_scatter_kernel<<<(N_TOK + 255) / 256, 256, 0, stream>>>(tk_idx, tk_w, cnt, list, lw);

    size_t smem_shared = (size_t)(16 * XS_STRIDE + 16 * HSH_STRIDE) * sizeof(float); // ~152.5 KB
    moe_shared_kernel<<<N_TOK / 16, 256, smem_shared, stream>>>(x, w1s, w2s, w3s, out);

    size_t smem_routed = (size_t)(16 * XS_STRIDE + 16 * HS_STRIDE) * sizeof(float);  // ~108.5 KB
    dim3 rgrid(E_NUM, N_TOK / 16);
    moe_routed_kernel<<<rgrid, 256, smem_routed, stream>>>(
        x, w1r, w2r, w3r, cnt, list, lw, out);
}